// UnifiedGNNLayer_47493748359696
// MI455X (gfx1250) — compile-verified
//
#include <hip/hip_runtime.h>
#include <math.h>

typedef __attribute__((ext_vector_type(2))) float v2f;
typedef __attribute__((ext_vector_type(8))) float v8f;

#define DD 128
#define HH 8
#define DHH 16
#define BN_EPS 1e-5f
#define LDST 132   // padded LDS row stride (floats): 528B -> conflict-free A reads

// ---------------------------------------------------------------------------
// float atomic max via monotonic int punning (init must be -inf)
__device__ inline void atomicMaxF(float* addr, float v) {
    if (v >= 0.0f) atomicMax((int*)addr, __float_as_int(v));
    else           atomicMin((unsigned int*)addr, __float_as_uint(v));
}

// ---------------------------------------------------------------------------
// Kernel 0: ea = merge MLP of concat(nt_emb, et_emb). One block.
__global__ __launch_bounds__(128)
void ea_kernel(const float* __restrict__ nt, const float* __restrict__ et,
               const float* __restrict__ mW1, const float* __restrict__ mb1,
               const float* __restrict__ m_g, const float* __restrict__ m_b,
               const float* __restrict__ m_m, const float* __restrict__ m_v,
               const float* __restrict__ mW2, const float* __restrict__ mb2,
               float* __restrict__ ea) {
    __shared__ float merged[2 * DD];
    __shared__ float hbuf[DD];
    int t = threadIdx.x;
    merged[t]      = nt[t];
    merged[DD + t] = et[t];
    __syncthreads();
    float acc = mb1[t];
    for (int i = 0; i < 2 * DD; ++i) acc += merged[i] * mW1[i * DD + t];
    acc = (acc - m_m[t]) * rsqrtf(m_v[t] + BN_EPS) * m_g[t] + m_b[t];
    hbuf[t] = fmaxf(acc, 0.0f);
    __syncthreads();
    float acc2 = mb2[t];
    for (int i = 0; i < DD; ++i) acc2 += hbuf[i] * mW2[i * DD + t];
    ea[t] = acc2;
}

// ---------------------------------------------------------------------------
// Init: smax=-inf, ssum=0, aggr=0
__global__ void init_kernel(float* __restrict__ smax, float* __restrict__ ssum,
                            float* __restrict__ aggr, int nSeg, long long nAg) {
    long long i = (long long)blockIdx.x * blockDim.x + threadIdx.x;
    if (i < nSeg) { smax[i] = -INFINITY; ssum[i] = 0.0f; }
    if (i < nAg)  aggr[i] = 0.0f;
}

// ---------------------------------------------------------------------------
// Weight repack into WMMA-native pairs: P[(kb*2+p)*128 + n] = {W[k][n], W[k+1][n]},
// k = kb*4 + 2p.  One global_load_b64 then supplies a lane's full B fragment.
__global__ __launch_bounds__(256)
void pack_kernel(const float* __restrict__ W, v2f* __restrict__ P) {
    int idx = blockIdx.x * 256 + threadIdx.x;   // 8192 per matrix
    int n = idx & 127;
    int pairIdx = idx >> 7;                     // 0..63
    int kb = pairIdx >> 1, p = pairIdx & 1;
    int k = kb * 4 + p * 2;
    v2f v; v.x = W[k * DD + n]; v.y = W[(k + 1) * DD + n];
    P[idx] = v;
}

// ---------------------------------------------------------------------------
// One 16x64 output tile pass: acc[4] covers 4 N-tiles over K=128 (32 wmma steps
// per tile). A from LDS (row = this lane's M row), B from packed global weights.
template<bool ADD_EA>
__device__ __forceinline__ void gemm_pass(const v2f* __restrict__ Wp,
                                          const float* __restrict__ arowp,
                                          const float* __restrict__ eas,
                                          int kh, int colbase, v8f acc[4]) {
    #pragma unroll
    for (int nt = 0; nt < 4; ++nt) acc[nt] = (v8f){0,0,0,0,0,0,0,0};
    const int ph = kh >> 1;  // which K-pair this lane half owns
    #pragma unroll 8
    for (int kb = 0; kb < 32; ++kb) {
        v2f Av = *(const v2f*)(arowp + kb * 4 + kh);
        if (ADD_EA) {
            v2f Ev = *(const v2f*)(eas + kb * 4 + kh);
            Av.x += Ev.x; Av.y += Ev.y;
        }
        const v2f* wp = Wp + (kb * 2 + ph) * DD + colbase;
        #pragma unroll
        for (int nt = 0; nt < 4; ++nt) {
            v2f Bv = wp[nt * 16];
            acc[nt] = __builtin_amdgcn_wmma_f32_16x16x4_f32(
                false, Av, false, Bv, (short)0, acc[nt], false, false);
        }
    }
}

template<bool HAS_BIAS>
__device__ __forceinline__ void store_tile(float* __restrict__ out,
                                           const float* __restrict__ bias, float scale,
                                           const v8f acc[4], int base, int rb,
                                           int colbase, int N) {
    #pragma unroll
    for (int nt = 0; nt < 4; ++nt) {
        int n = colbase + nt * 16;
        float b = HAS_BIAS ? bias[n] : 0.0f;
        #pragma unroll
        for (int j = 0; j < 8; ++j) {
            int node = base + rb + j;
            if (node < N) out[(size_t)node * DD + n] = (acc[nt][j] + b) * scale;
        }
    }
}

// ---------------------------------------------------------------------------
// Kernel 1: per-node GEMMs with f32 WMMA.
//   Q = ((x+ea) @ qW + qb) * 0.25 ; K = (x+ea) @ kW + kb
//   V = x @ vW + vb               ; R = x @ resW
// Block: 256 thr (8 waves), 64 nodes. Wave w: mtile=w%4, nhalf=w/4.
__global__ __launch_bounds__(256)
void node_qkvr_kernel(const float* __restrict__ x, const float* __restrict__ ea,
                      const v2f* __restrict__ qWp, const float* __restrict__ qb,
                      const v2f* __restrict__ kWp, const float* __restrict__ kb,
                      const v2f* __restrict__ vWp, const float* __restrict__ vb,
                      const v2f* __restrict__ rWp,
                      float* __restrict__ Q, float* __restrict__ K,
                      float* __restrict__ V, float* __restrict__ R, int N) {
    __shared__ float xs[64 * LDST];
    __shared__ float eas[DD];
    const int t = threadIdx.x;
    const int base = blockIdx.x * 64;

    // stage x tile (64x128 f32) via float4
    #pragma unroll
    for (int j = 0; j < 8; ++j) {
        int u = t + j * 256;            // float4 unit id (2048 total)
        int row = u >> 5, c4 = u & 31;
        int node = base + row; if (node >= N) node = N - 1;
        float4 xv = ((const float4*)x)[(size_t)node * 32 + c4];
        ((float4*)&xs[row * LDST])[c4] = xv;
    }
    if (t < DD) eas[t] = ea[t];
    __syncthreads();

    const int wave = t >> 5, lane = t & 31;
    const int mtile = wave & 3, nhalf = wave >> 2;
    const int l16 = lane & 15;
    const int kh = (lane >> 4) << 1;          // K pair select: 0 or 2
    const float* arowp = &xs[(mtile * 16 + l16) * LDST];
    const int colbase = nhalf * 64 + l16;
    const int rb = mtile * 16 + ((lane >> 4) << 3);

    v8f acc[4];
    gemm_pass<true >(qWp, arowp, eas, kh, colbase, acc);
    store_tile<true >(Q, qb, 0.25f, acc, base, rb, colbase, N);
    gemm_pass<true >(kWp, arowp, eas, kh, colbase, acc);
    store_tile<true >(K, kb, 1.0f, acc, base, rb, colbase, N);
    gemm_pass<false>(vWp, arowp, eas, kh, colbase, acc);
    store_tile<true >(V, vb, 1.0f, acc, base, rb, colbase, N);
    gemm_pass<false>(rWp, arowp, eas, kh, colbase, acc);
    store_tile<false>(R, nullptr, 1.0f, acc, base, rb, colbase, N);
}

// ---------------------------------------------------------------------------
// Kernel 2: per-(edge,head) attention scores + segment max (by src)
__global__ __launch_bounds__(256)
void edge_score_kernel(const int* __restrict__ ei, const float* __restrict__ Q,
                       const float* __restrict__ K, float* __restrict__ S,
                       float* __restrict__ smax, int E) {
    long long idx = (long long)blockIdx.x * blockDim.x + threadIdx.x;
    if (idx >= (long long)E * HH) return;
    int e = (int)(idx >> 3), h = (int)(idx & 7);
    int s = ei[e], d = ei[E + e];
    const float4* q = (const float4*)(Q + (size_t)s * DD + h * DHH);
    const float4* k = (const float4*)(K + (size_t)d * DD + h * DHH);
    float acc = 0.0f;
    #pragma unroll
    for (int j = 0; j < 4; ++j) {
        float4 a = q[j], b = k[j];
        acc += a.x * b.x + a.y * b.y + a.z * b.z + a.w * b.w;
    }
    S[idx] = acc;
    atomicMaxF(&smax[s * HH + h], acc);
}

// ---------------------------------------------------------------------------
// Kernel 3: ex = exp(s - smax[src]); segment sum (by src)
__global__ __launch_bounds__(256)
void edge_exp_kernel(const int* __restrict__ ei, float* __restrict__ S,
                     const float* __restrict__ smax, float* __restrict__ ssum, int E) {
    long long idx = (long long)blockIdx.x * blockDim.x + threadIdx.x;
    if (idx >= (long long)E * HH) return;
    int e = (int)(idx >> 3), h = (int)(idx & 7);
    int s = ei[e];
    float ex = expf(S[idx] - smax[s * HH + h]);
    S[idx] = ex;
    atomicAdd(&ssum[s * HH + h], ex);
}

// ---------------------------------------------------------------------------
// Kernel 4: alpha = ex/ssum[src]; aggr[dst] += V[src]*alpha (per 4-float chunk)
__global__ __launch_bounds__(256)
void edge_aggr_kernel(const int* __restrict__ ei, const float* __restrict__ S,
                      const float* __restrict__ ssum, const float* __restrict__ V,
                      float* __restrict__ aggr, int E) {
    long long idx = (long long)blockIdx.x * blockDim.x + threadIdx.x;
    if (idx >= (long long)E * 32) return;
    int e = (int)(idx >> 5), q = (int)(idx & 31);
    int h = q >> 2;
    int s = ei[e], d = ei[E + e];
    float alpha = S[(size_t)e * HH + h] / ssum[s * HH + h];
    float4 v = ((const float4*)V)[(size_t)s * 32 + q];
    float* dst = aggr + (size_t)d * DD + q * 4;
    atomicAdd(dst + 0, v.x * alpha);
    atomicAdd(dst + 1, v.y * alpha);
    atomicAdd(dst + 2, v.z * alpha);
    atomicAdd(dst + 3, v.w * alpha);
}

// ---------------------------------------------------------------------------
// Kernel 5: z = aggr + res; out = leaky_relu(BN(z@oW1+ob1)) @ oW2 + ob2
__global__ __launch_bounds__(256)
void out_mlp_kernel(const float* __restrict__ aggr, const float* __restrict__ res,
                    const v2f* __restrict__ oW1p, const float* __restrict__ ob1,
                    const float* __restrict__ o_g, const float* __restrict__ o_b,
                    const float* __restrict__ o_m, const float* __restrict__ o_v,
                    const v2f* __restrict__ oW2p, const float* __restrict__ ob2,
                    float* __restrict__ out, int N) {
    __shared__ float zs[64 * LDST];
    const int t = threadIdx.x;
    const int base = blockIdx.x * 64;

    #pragma unroll
    for (int j = 0; j < 8; ++j) {
        int u = t + j * 256;
        int row = u >> 5, c4 = u & 31;
        int node = base + row; if (node >= N) node = N - 1;
        float4 a = ((const float4*)aggr)[(size_t)node * 32 + c4];
        float4 r = ((const float4*)res)[(size_t)node * 32 + c4];
        float4 z; z.x = a.x + r.x; z.y = a.y + r.y; z.z = a.z + r.z; z.w = a.w + r.w;
        ((float4*)&zs[row * LDST])[c4] = z;
    }
    __syncthreads();

    const int wave = t >> 5, lane = t & 31;
    const int mtile = wave & 3, nhalf = wave >> 2;
    const int l16 = lane & 15;
    const int kh = (lane >> 4) << 1;
    const float* arowp = &zs[(mtile * 16 + l16) * LDST];
    const int colbase = nhalf * 64 + l16;
    const int rb = mtile * 16 + ((lane >> 4) << 3);

    // GEMM 1: zs @ oW1, then BN + leaky-relu into registers
    v8f acc[4];
    gemm_pass<false>(oW1p, arowp, nullptr, kh, colbase, acc);
    float uval[4][8];
    #pragma unroll
    for (int nt = 0; nt < 4; ++nt) {
        int n = colbase + nt * 16;
        float bias = ob1[n];
        float rs = rsqrtf(o_v[n] + BN_EPS) * o_g[n];
        float mn = o_m[n], bb = o_b[n];
        #pragma unroll
        for (int j = 0; j < 8; ++j) {
            float tv = (acc[nt][j] + bias - mn) * rs + bb;
            uval[nt][j] = tv > 0.0f ? tv : 0.01f * tv;
        }
    }
    __syncthreads();   // everyone done reading zs
    #pragma unroll
    for (int nt = 0; nt < 4; ++nt) {
        int n = colbase + nt * 16;
        #pragma unroll
        for (int j = 0; j < 8; ++j) zs[(rb + j) * LDST + n] = uval[nt][j];
    }
    __syncthreads();

    // GEMM 2: u @ oW2
    gemm_pass<false>(oW2p, arowp, nullptr, kh, colbase, acc);
    store_tile<true>(out, ob2, 1.0f, acc, base, rb, colbase, N);
}

// ---------------------------------------------------------------------------
extern "C" void kernel_launch(void* const* d_in, const int* in_sizes, int n_in,
                              void* d_out, int out_size, void* d_ws, size_t ws_size,
                              hipStream_t stream) {
    (void)n_in; (void)out_size; (void)ws_size;
    const float* x    = (const float*)d_in[0];
    const int*   ei   = (const int*)  d_in[1];
    const float* nt   = (const float*)d_in[2];
    const float* et   = (const float*)d_in[3];
    const float* mW1  = (const float*)d_in[4];
    const float* mb1  = (const float*)d_in[5];
    const float* m_g  = (const float*)d_in[6];
    const float* m_b  = (const float*)d_in[7];
    const float* m_m  = (const float*)d_in[8];
    const float* m_v  = (const float*)d_in[9];
    const float* mW2  = (const float*)d_in[10];
    const float* mb2  = (const float*)d_in[11];
    const float* resW = (const float*)d_in[12];
    const float* qW   = (const float*)d_in[13];
    const float* qb   = (const float*)d_in[14];
    const float* kW   = (const float*)d_in[15];
    const float* kb   = (const float*)d_in[16];
    const float* vW   = (const float*)d_in[17];
    const float* vb   = (const float*)d_in[18];
    const float* oW1  = (const float*)d_in[19];
    const float* ob1  = (const float*)d_in[20];
    const float* o_g  = (const float*)d_in[21];
    const float* o_b  = (const float*)d_in[22];
    const float* o_m  = (const float*)d_in[23];
    const float* o_v  = (const float*)d_in[24];
    const float* oW2  = (const float*)d_in[25];
    const float* ob2  = (const float*)d_in[26];
    float* out = (float*)d_out;

    const int N = in_sizes[0] / DD;
    const int E = in_sizes[1] / 2;
    const size_t NM = (size_t)N * DD;
    const size_t PACKF = 2 * 8192;   // floats per packed matrix

    float* w    = (float*)d_ws;
    float* ea   = w;  w += 256;
    float* Q    = w;  w += NM;
    float* K    = w;  w += NM;
    float* V    = w;  w += NM;
    float* R    = w;  w += NM;
    float* aggr = w;  w += NM;
    float* S    = w;  w += (size_t)E * HH;
    float* smax = w;  w += (size_t)N * HH;
    float* ssum = w;  w += (size_t)N * HH;
    v2f* qWp  = (v2f*)w;  w += PACKF;
    v2f* kWp  = (v2f*)w;  w += PACKF;
    v2f* vWp  = (v2f*)w;  w += PACKF;
    v2f* rWp  = (v2f*)w;  w += PACKF;
    v2f* oW1p = (v2f*)w;  w += PACKF;
    v2f* oW2p = (v2f*)w;  w += PACKF;

    ea_kernel<<<1, 128, 0, stream>>>(nt, et, mW1, mb1, m_g, m_b, m_m, m_v, mW2, mb2, ea);

    long long nAg = (long long)NM;
    int initBlocks = (int)((nAg + 255) / 256);
    init_kernel<<<initBlocks, 256, 0, stream>>>(smax, ssum, aggr, N * HH, nAg);

    pack_kernel<<<32, 256, 0, stream>>>(qW,   qWp);
    pack_kernel<<<32, 256, 0, stream>>>(kW,   kWp);
    pack_kernel<<<32, 256, 0, stream>>>(vW,   vWp);
    pack_kernel<<<32, 256, 0, stream>>>(resW, rWp);
    pack_kernel<<<32, 256, 0, stream>>>(oW1,  oW1p);
    pack_kernel<<<32, 256, 0, stream>>>(oW2,  oW2p);

    int nodeBlocks = (N + 63) / 64;
    node_qkvr_kernel<<<nodeBlocks, 256, 0, stream>>>(
        x, ea, qWp, qb, kWp, kb, vWp, vb, rWp, Q, K, V, R, N);

    long long eh = (long long)E * HH;
    edge_score_kernel<<<(int)((eh + 255) / 256), 256, 0, stream>>>(ei, Q, K, S, smax, E);
    edge_exp_kernel<<<(int)((eh + 255) / 256), 256, 0, stream>>>(ei, S, smax, ssum, E);

    long long e32 = (long long)E * 32;
    edge_aggr_kernel<<<(int)((e32 + 255) / 256), 256, 0, stream>>>(ei, S, ssum, V, aggr, E);

    out_mlp_kernel<<<nodeBlocks, 256, 0, stream>>>(
        aggr, R, oW1p, ob1, o_g, o_b, o_m, o_v, oW2p, ob2, out, N);
}